// MapDensePoseTexModule_22101901705512
// MI455X (gfx1250) — compile-verified
//
#include <hip/hip_runtime.h>

// MapDensePoseTex gather kernel for MI455X (gfx1250).
// Pure gather/scatter workload: no dense matmul -> WMMA/TDM not applicable.
// Strategy: exploit 192MB L2 (tex 128MiB + lut 12MiB stay resident) by
// streaming iuv/out with non-temporal cache hints (th:TH_LOAD_NT /
// th:TH_STORE_NT); all 3 iuv plane loads issued in one clause up front
// (single load round-trip); 16-wide unrolled gather for MLP;
// wave32-coalesced planar loads/stores.

#define HW_   262144   // 512*512 pixels per image
#define C_CH  16       // tex channels
#define B_N   8        // batch

__global__ __launch_bounds__(256) void mapdensepose_gather_kernel(
    const float* __restrict__ tex,
    const int*   __restrict__ iuv,
    const float* __restrict__ lut,
    const int*   __restrict__ tex_res_p,
    float*       __restrict__ out)
{
    const int t  = blockIdx.x * blockDim.x + threadIdx.x;  // [0, B*H*W)
    const int b  = t >> 18;          // t / HW_
    const int yx = t & (HW_ - 1);    // t % HW_

    // iuv is planar: [b][plane][y][x]; consecutive threads -> consecutive x
    // -> fully coalesced. Issue all three plane loads unconditionally and
    // back-to-back (one clause, one wait) -- ~96% of lanes need u/v anyway,
    // and this removes a serialized second memory round-trip.
    // Streamed once: non-temporal loads keep the 24MiB of iuv out of L2.
    const int* iuv_b = iuv + (size_t)b * 3 * HW_ + yx;
    const int part = __builtin_nontemporal_load(iuv_b);
    const int xu   = __builtin_nontemporal_load(iuv_b + HW_);
    const int xv   = __builtin_nontemporal_load(iuv_b + 2 * HW_);
    const int R    = *tex_res_p;     // uniform scalar (512)

    // Branch-independent index math (replicates reference f32 arithmetic
    // exactly; jnp.round == rintf == round-half-to-even).
    const float u = fminf(fmaxf((float)xu * (1.0f / 255.0f), 0.0f), 1.0f);
    const float v = fminf(fmaxf((float)xv * (1.0f / 255.0f), 0.0f), 1.0f);
    const int ui = (int)rintf(u * 255.0f);
    const int vi = (int)rintf(v * 255.0f);
    const int pi = min(max(part - 1, 0), 23);

    float vals[C_CH];
    if (part > 0) {
        // lut[pi][vi][ui] as one 8B load (default RT policy: 12MiB L2-resident)
        const float2 uvs = ((const float2*)lut)[((pi << 8) | vi) * 256 + ui];

        const float Rm1 = (float)(R - 1);
        const int u_I = (int)rintf(uvs.x * Rm1);
        const int v_I = (int)rintf((1.0f - uvs.y) * Rm1);

        const size_t cstride = (size_t)R * (size_t)R;
        const float* tb = tex + (size_t)b * C_CH * cstride
                              + (size_t)v_I * (size_t)R + (size_t)u_I;
        // 16 independent scattered 4B loads in flight (MLP); tex is 128MiB
        // and L2-resident on MI455X, so these are mostly L2 hits.
        #pragma unroll
        for (int c = 0; c < C_CH; ++c)
            vals[c] = tb[(size_t)c * cstride];
    } else {
        #pragma unroll
        for (int c = 0; c < C_CH; ++c) vals[c] = 0.0f;
    }

    // out[b][c][y][x]: per-channel plane, coalesced across the wave.
    // Streamed once -> non-temporal stores (th:TH_STORE_NT) protect L2.
    float* ob = out + (size_t)b * C_CH * HW_ + yx;
    #pragma unroll
    for (int c = 0; c < C_CH; ++c)
        __builtin_nontemporal_store(vals[c], ob + (size_t)c * HW_);
}

extern "C" void kernel_launch(void* const* d_in, const int* in_sizes, int n_in,
                              void* d_out, int out_size, void* d_ws, size_t ws_size,
                              hipStream_t stream) {
    const float* tex     = (const float*)d_in[0];
    const int*   iuv     = (const int*)d_in[1];
    const float* lut     = (const float*)d_in[2];
    const int*   tex_res = (const int*)d_in[3];
    float*       out     = (float*)d_out;

    const int total  = B_N * HW_;        // 2,097,152 pixels
    const int block  = 256;              // 8 wave32s per workgroup
    const int blocks = total / block;    // divides evenly (8192)

    mapdensepose_gather_kernel<<<blocks, block, 0, stream>>>(
        tex, iuv, lut, tex_res, out);
}